// SAGE_5222680232229
// MI455X (gfx1250) — compile-verified
//
#include <hip/hip_runtime.h>
#include <hip/hip_bf16.h>

// ---------------------------------------------------------------------------
// GraphSAGE (3x SAGEConv mean-aggr + global_add_pool + linear) for gfx1250.
// Dense GEMM parts use V_WMMA_F32_16X16X4_F32 (full f32 precision, K%4==0
// for both K=108 and K=88). Scatter/gather parts are edge-parallel float
// atomics (memory-bound; 23.3 TB/s HBM is the limiter, weights live in L2).
// GEMM blocking: one wave owns a 16-row M-panel and ALL column tiles, so the
// strided A fragment is loaded once per K-step and reused across 6-7 WMMAs.
// ---------------------------------------------------------------------------

#define N_NODES 50000
#define N_EDGES 800000
#define N_GRAPHS 256
#define C_FEAT 108
#define MLP_DIM 88
#define OUT_DIM 100

typedef __attribute__((ext_vector_type(2))) float v2f;
typedef __attribute__((ext_vector_type(8))) float v8f;

static inline unsigned cdiv(long a, long b) { return (unsigned)((a + b - 1) / b); }

// ---------------------------------------------------------------------------
// Degree count: one thread per edge, atomic float add into cnt[dst].
// ---------------------------------------------------------------------------
__global__ void sage_degree_kernel(const int* __restrict__ dst, float* __restrict__ cnt, int E) {
    int e = blockIdx.x * blockDim.x + threadIdx.x;
    if (e < E) atomicAdd(&cnt[dst[e]], 1.0f);
}

// cnt[i] -> 1 / max(cnt[i], 1)
__global__ void sage_invdeg_kernel(float* __restrict__ cnt, int n) {
    int i = blockIdx.x * blockDim.x + threadIdx.x;
    if (i < n) cnt[i] = 1.0f / fmaxf(cnt[i], 1.0f);
}

// ---------------------------------------------------------------------------
// Zero-pad a KxN weight (row-major) into a Kxldw buffer (cols >= N zeroed).
// ---------------------------------------------------------------------------
__global__ void sage_padw_kernel(const float* __restrict__ W, float* __restrict__ Wp,
                                 int K, int N, int ldw) {
    int t = blockIdx.x * blockDim.x + threadIdx.x;
    int total = K * ldw;
    if (t >= total) return;
    int k = t / ldw, n = t % ldw;
    Wp[t] = (n < N) ? W[k * N + n] : 0.0f;
}

// ---------------------------------------------------------------------------
// Edge-parallel scatter-add: agg[dst[e], :] += x[src[e], :]. One thread per
// (edge, float4-chunk); consecutive threads cover one row -> coalesced reads.
// ---------------------------------------------------------------------------
__global__ void sage_scatter_add_kernel(const int* __restrict__ src, const int* __restrict__ dst,
                                        const float* __restrict__ x, float* __restrict__ agg,
                                        int E, int D) {
    int chunks = D >> 2;
    long t = (long)blockIdx.x * blockDim.x + threadIdx.x;
    long total = (long)E * chunks;
    if (t >= total) return;
    int e = (int)(t / chunks);
    int c = (int)(t % chunks);
    int s = src[e], d = dst[e];
    const float4 v = *(const float4*)(x + (long)s * D + 4 * c);
    float* base = agg + (long)d * D + 4 * c;
    atomicAdd(base + 0, v.x);
    atomicAdd(base + 1, v.y);
    atomicAdd(base + 2, v.z);
    atomicAdd(base + 3, v.w);
}

// ---------------------------------------------------------------------------
// Node-parallel pooled[batch[n], :] += h[n, :] (global_add_pool).
// ---------------------------------------------------------------------------
__global__ void sage_pool_kernel(const int* __restrict__ batch, const float* __restrict__ h,
                                 float* __restrict__ pooled, int Nn, int D) {
    int chunks = D >> 2;
    long t = (long)blockIdx.x * blockDim.x + threadIdx.x;
    long total = (long)Nn * chunks;
    if (t >= total) return;
    int n = (int)(t / chunks);
    int c = (int)(t % chunks);
    int b = batch[n];
    const float4 v = *(const float4*)(h + (long)n * D + 4 * c);
    float* base = pooled + (long)b * D + 4 * c;
    atomicAdd(base + 0, v.x);
    atomicAdd(base + 1, v.y);
    atomicAdd(base + 2, v.z);
    atomicAdd(base + 3, v.w);
}

// ---------------------------------------------------------------------------
// WMMA f32 GEMM: out = relu?( [scale1 .* A1] @ B1  (+ A2 @ B2)  + bias )
//   A1, A2 : [M x K] row-major (K % 4 == 0, M % 16 == 0)
//   scale1 : per-row scale applied to A1 (mean-divide folded in), or null
//   B1, B2 : [K x NT*16] row-major, zero-padded so every 16-col tile is valid
//   out    : [M x Nvalid] row-major; columns >= Nvalid are discarded
// One wave per 16-row M-panel, computing all NT column tiles: the strided
// A fragment is fetched once per K-step and reused across NT independent
// V_WMMA_F32_16X16X4_F32 accumulations (no inter-WMMA RAW hazards).
// A-frag layout (ISA 7.12.2): lane 0-15 -> K=(k,k+1), lane 16-31 -> K=(k+2,k+3),
// row = m0 + (lane&15). C/D: VGPR r -> row m0 + r + 8*(lane>=16), col n0+(lane&15).
// ---------------------------------------------------------------------------
template <int NT>
__global__ void sage_wmma_gemm_kernel(const float* __restrict__ A1, const float* __restrict__ scale1,
                                      const float* __restrict__ A2,
                                      const float* __restrict__ B1, const float* __restrict__ B2,
                                      const float* __restrict__ bias, float* __restrict__ out,
                                      int M, int K, int Nvalid, int doRelu) {
    const int ldb = NT * 16;
    int wave = blockIdx.x * (blockDim.x >> 5) + (threadIdx.x >> 5);
    int Mtiles = M >> 4;
    if (wave >= Mtiles) return;  // uniform per-wave exit, before any WMMA

    int m0 = wave << 4;
    int lane = threadIdx.x & 31;
    int half = lane >> 4;   // 0: K pair (k,k+1) | 1: K pair (k+2,k+3)
    int l15  = lane & 15;
    int mrow = m0 + l15;

    v8f acc[NT];
    #pragma unroll
    for (int nt = 0; nt < NT; ++nt) acc[nt] = (v8f){0.f, 0.f, 0.f, 0.f, 0.f, 0.f, 0.f, 0.f};

    // ---- term 1: (scale1 .* A1) @ B1 ----
    {
        float s = scale1 ? scale1[mrow] : 1.0f;
        const float* aptr = A1 + (long)mrow * K;
        for (int k = 0; k < K; k += 4) {
            v2f a = *(const v2f*)(aptr + k + 2 * half);
            a *= s;
            const float* brow0 = B1 + (long)(k + 2 * half + 0) * ldb + l15;
            const float* brow1 = B1 + (long)(k + 2 * half + 1) * ldb + l15;
            #pragma unroll
            for (int nt = 0; nt < NT; ++nt) {
                v2f b;
                b.x = brow0[nt * 16];
                b.y = brow1[nt * 16];
                acc[nt] = __builtin_amdgcn_wmma_f32_16x16x4_f32(false, a, false, b,
                                                                (short)0, acc[nt], false, false);
            }
        }
    }

    // ---- term 2 (optional): A2 @ B2 ----
    if (A2) {
        const float* aptr = A2 + (long)mrow * K;
        for (int k = 0; k < K; k += 4) {
            v2f a = *(const v2f*)(aptr + k + 2 * half);
            const float* brow0 = B2 + (long)(k + 2 * half + 0) * ldb + l15;
            const float* brow1 = B2 + (long)(k + 2 * half + 1) * ldb + l15;
            #pragma unroll
            for (int nt = 0; nt < NT; ++nt) {
                v2f b;
                b.x = brow0[nt * 16];
                b.y = brow1[nt * 16];
                acc[nt] = __builtin_amdgcn_wmma_f32_16x16x4_f32(false, a, false, b,
                                                                (short)0, acc[nt], false, false);
            }
        }
    }

    // ---- epilogue: bias (+ relu), guarded store ----
    #pragma unroll
    for (int nt = 0; nt < NT; ++nt) {
        int n = nt * 16 + l15;
        if (n < Nvalid) {
            float bv = bias[n];
            #pragma unroll
            for (int r = 0; r < 8; ++r) {
                int m = m0 + r + 8 * half;
                float v = acc[nt][r] + bv;
                if (doRelu) v = fmaxf(v, 0.0f);
                out[(long)m * Nvalid + n] = v;
            }
        }
    }
}

// ---------------------------------------------------------------------------
// Host-side launch orchestration (graph-capture safe: only async memsets and
// kernel launches on `stream`).
// ---------------------------------------------------------------------------
extern "C" void kernel_launch(void* const* d_in, const int* in_sizes, int n_in,
                              void* d_out, int out_size, void* d_ws, size_t ws_size,
                              hipStream_t stream) {
    (void)in_sizes; (void)n_in; (void)out_size; (void)ws_size;

    const float* x    = (const float*)d_in[0];
    const int*   ei   = (const int*)d_in[1];   // [2, E]
    const int*   bat  = (const int*)d_in[2];
    const float* W1l  = (const float*)d_in[3];
    const float* W1r  = (const float*)d_in[4];
    const float* b1   = (const float*)d_in[5];
    const float* W2l  = (const float*)d_in[6];
    const float* W2r  = (const float*)d_in[7];
    const float* b2   = (const float*)d_in[8];
    const float* W3l  = (const float*)d_in[9];
    const float* W3r  = (const float*)d_in[10];
    const float* b3   = (const float*)d_in[11];
    const float* Wlin = (const float*)d_in[12];
    const float* blin = (const float*)d_in[13];
    float* out = (float*)d_out;

    const int* src = ei;
    const int* dst = ei + N_EDGES;

    const int LDW  = 96;   // 88 -> 6 tiles of 16
    const int LDWF = 112;  // 100 -> 7 tiles of 16

    // Workspace carve-up (floats)
    float* w = (float*)d_ws;
    float* inv    = w;                           // N_NODES          (degree -> 1/max(deg,1))
    float* agg    = inv    + N_NODES;            // N_NODES * C_FEAT (reused at width 88 too)
    float* h0     = agg    + (long)N_NODES * C_FEAT;     // N_NODES * MLP_DIM
    float* h1     = h0     + (long)N_NODES * MLP_DIM;    // N_NODES * MLP_DIM
    float* W1lp   = h1     + (long)N_NODES * MLP_DIM;    // C_FEAT * LDW
    float* W1rp   = W1lp   + C_FEAT * LDW;
    float* W2lp   = W1rp   + C_FEAT * LDW;               // MLP_DIM * LDW
    float* W2rp   = W2lp   + MLP_DIM * LDW;
    float* W3lp   = W2rp   + MLP_DIM * LDW;
    float* W3rp   = W3lp   + MLP_DIM * LDW;
    float* Wlinp  = W3rp   + MLP_DIM * LDW;              // MLP_DIM * LDWF
    float* pooled = Wlinp  + MLP_DIM * LDWF;             // N_GRAPHS * MLP_DIM

    const int T = 256;

    // --- degrees ---
    hipMemsetAsync(inv, 0, sizeof(float) * N_NODES, stream);
    sage_degree_kernel<<<cdiv(N_EDGES, T), T, 0, stream>>>(dst, inv, N_EDGES);
    sage_invdeg_kernel<<<cdiv(N_NODES, T), T, 0, stream>>>(inv, N_NODES);

    // --- padded weights ---
    sage_padw_kernel<<<cdiv((long)C_FEAT * LDW, T), T, 0, stream>>>(W1l, W1lp, C_FEAT, MLP_DIM, LDW);
    sage_padw_kernel<<<cdiv((long)C_FEAT * LDW, T), T, 0, stream>>>(W1r, W1rp, C_FEAT, MLP_DIM, LDW);
    sage_padw_kernel<<<cdiv((long)MLP_DIM * LDW, T), T, 0, stream>>>(W2l, W2lp, MLP_DIM, MLP_DIM, LDW);
    sage_padw_kernel<<<cdiv((long)MLP_DIM * LDW, T), T, 0, stream>>>(W2r, W2rp, MLP_DIM, MLP_DIM, LDW);
    sage_padw_kernel<<<cdiv((long)MLP_DIM * LDW, T), T, 0, stream>>>(W3l, W3lp, MLP_DIM, MLP_DIM, LDW);
    sage_padw_kernel<<<cdiv((long)MLP_DIM * LDW, T), T, 0, stream>>>(W3r, W3rp, MLP_DIM, MLP_DIM, LDW);
    sage_padw_kernel<<<cdiv((long)MLP_DIM * LDWF, T), T, 0, stream>>>(Wlin, Wlinp, MLP_DIM, OUT_DIM, LDWF);

    const unsigned gemmBlocks = cdiv(N_NODES / 16, 4);  // 1 wave per 16-row panel, 4 waves/block

    // --- layer 1: agg = scatter-mean(x); h0 = relu(agg@W1l + x@W1r + b1) ---
    hipMemsetAsync(agg, 0, sizeof(float) * (long)N_NODES * C_FEAT, stream);
    sage_scatter_add_kernel<<<cdiv((long)N_EDGES * (C_FEAT / 4), T), T, 0, stream>>>(
        src, dst, x, agg, N_EDGES, C_FEAT);
    sage_wmma_gemm_kernel<6><<<gemmBlocks, 128, 0, stream>>>(
        agg, inv, x, W1lp, W1rp, b1, h0, N_NODES, C_FEAT, MLP_DIM, 1);

    // --- layer 2 ---
    hipMemsetAsync(agg, 0, sizeof(float) * (long)N_NODES * MLP_DIM, stream);
    sage_scatter_add_kernel<<<cdiv((long)N_EDGES * (MLP_DIM / 4), T), T, 0, stream>>>(
        src, dst, h0, agg, N_EDGES, MLP_DIM);
    sage_wmma_gemm_kernel<6><<<gemmBlocks, 128, 0, stream>>>(
        agg, inv, h0, W2lp, W2rp, b2, h1, N_NODES, MLP_DIM, MLP_DIM, 1);

    // --- layer 3 ---
    hipMemsetAsync(agg, 0, sizeof(float) * (long)N_NODES * MLP_DIM, stream);
    sage_scatter_add_kernel<<<cdiv((long)N_EDGES * (MLP_DIM / 4), T), T, 0, stream>>>(
        src, dst, h1, agg, N_EDGES, MLP_DIM);
    sage_wmma_gemm_kernel<6><<<gemmBlocks, 128, 0, stream>>>(
        agg, inv, h1, W3lp, W3rp, b3, h0, N_NODES, MLP_DIM, MLP_DIM, 1);

    // --- global_add_pool ---
    hipMemsetAsync(pooled, 0, sizeof(float) * (long)N_GRAPHS * MLP_DIM, stream);
    sage_pool_kernel<<<cdiv((long)N_NODES * (MLP_DIM / 4), T), T, 0, stream>>>(
        bat, h0, pooled, N_NODES, MLP_DIM);

    // --- final linear: out = pooled @ Wlin + blin  (256x88 @ 88x100) ---
    sage_wmma_gemm_kernel<7><<<cdiv(N_GRAPHS / 16, 4), 128, 0, stream>>>(
        pooled, nullptr, nullptr, Wlinp, nullptr, blin, out,
        N_GRAPHS, MLP_DIM, OUT_DIM, 0);
}